// DAttention_15135464751187
// MI455X (gfx1250) — compile-verified
//
#include <hip/hip_runtime.h>
#include <hip/hip_bf16.h>

// CDNA5 / gfx1250. wave32. Two-pass bandwidth-bound kernel:
//   pass 1: davg[b,d] = mean over (C,H,W)  (WMMA-assisted wave reduction)
//   pass 2: out = x * davg[b,d]            (uniform-per-block scale, NT stores)

typedef float v2f __attribute__((ext_vector_type(2)));
typedef float v4f __attribute__((ext_vector_type(4)));
typedef float v8f __attribute__((ext_vector_type(8)));

// Problem dims (fixed by the reference): B=8, C=32, D=32, H=128, W=128.
#define NB 8
#define NC 32
#define ND 32
#define HW 16384u              // 128*128 elements per (b,c,d) chunk
#define CHUNK_F4 4096u         // HW/4 float4 per chunk
#define N_CHUNKS 8192u         // B*C*D chunks
#define INV_CHW (1.0f / 524288.0f)   // 1 / (C*H*W)

// Full 32-lane wave sum using V_WMMA_F32_16X16X4_F32 with B = ones.
// A layout (ISA 7.12.2, 32-bit A 16x4): lanes 0-15 -> {K0,K1}, lanes 16-31 -> {K2,K3},
// M = lane%16. Putting the partial in component 0 and zero in component 1 gives
// D[m][n] = p_m + p_{m+16}. Each lane sums its 8 D regs (rows 0-7 or 8-15),
// then one xor-16 shuffle completes the 32-lane total on every lane.
__device__ __forceinline__ float wave_reduce_wmma(float p) {
    v2f a;  a[0] = p;    a[1] = 0.0f;
    v2f bo; bo[0] = 1.0f; bo[1] = 1.0f;
    v8f c = {};
    v8f d = __builtin_amdgcn_wmma_f32_16x16x4_f32(
        /*neg_a=*/false, a, /*neg_b=*/false, bo,
        /*c_mod=*/(short)0, c, /*reuse_a=*/false, /*reuse_b=*/false);
    float r = ((d[0] + d[1]) + (d[2] + d[3])) + ((d[4] + d[5]) + (d[6] + d[7]));
    r += __shfl_xor(r, 16, 32);   // add rows 0-7 half with rows 8-15 half
    return r;
}

// Pass 1: one block per (b,d) pair (256 blocks x 1024 threads = 32 waves).
// Sums 32 contiguous 64KB chunks with coalesced float4 loads.
__global__ void __launch_bounds__(1024)
davg_kernel(const float* __restrict__ x, float* __restrict__ davg) {
    const unsigned bd  = blockIdx.x;           // 0..255
    const unsigned b   = bd >> 5;
    const unsigned d   = bd & 31;
    const unsigned tid = threadIdx.x;          // 0..1023

    float sum = 0.0f;
    #pragma unroll 2
    for (unsigned c = 0; c < NC; ++c) {
        // chunk id = (b*32 + c)*32 + d ; each chunk is HW contiguous floats
        const size_t chunk = (size_t)((b * NC + c) * ND + d);
        const v4f* __restrict__ p = (const v4f*)(x + (chunk << 14));
        #pragma unroll
        for (unsigned i = tid; i < CHUNK_F4; i += 1024) {
            v4f v = p[i];
            sum += (v[0] + v[1]) + (v[2] + v[3]);
        }
    }

    // wave32 reduce via WMMA, then cross-wave reduce via LDS.
    float r = wave_reduce_wmma(sum);

    __shared__ float wsum[32];
    const unsigned wave = tid >> 5;
    const unsigned lane = tid & 31;
    if (lane == 0) wsum[wave] = r;
    __syncthreads();
    if (tid == 0) {
        float t = 0.0f;
        #pragma unroll
        for (int w = 0; w < 32; ++w) t += wsum[w];
        davg[bd] = t * INV_CHW;
    }
}

// Pass 2: one block per 16384-element chunk (8192 blocks x 256 threads,
// 16 float4 per thread). Scale is uniform per block -> scalar load.
// Output is streamed with non-temporal stores (never re-read).
__global__ void __launch_bounds__(256)
scale_kernel(const float* __restrict__ x, const float* __restrict__ davg,
             float* __restrict__ out) {
    const unsigned chunk = blockIdx.x;                          // (b*32+c)*32+d
    const unsigned si    = ((chunk >> 10) << 5) | (chunk & 31); // b*32 + d
    const float    s     = davg[si];                            // uniform -> SGPR

    const v4f* __restrict__ xi = (const v4f*)x;
    v4f* __restrict__       oi = (v4f*)out;
    const size_t base = ((size_t)chunk << 12) + threadIdx.x;    // float4 index

    #pragma unroll
    for (int j = 0; j < 16; ++j) {
        const size_t k = base + (size_t)j * 256;
        v4f v = xi[k];
        v[0] *= s; v[1] *= s; v[2] *= s; v[3] *= s;
        __builtin_nontemporal_store(v, &oi[k]);
    }
}

extern "C" void kernel_launch(void* const* d_in, const int* in_sizes, int n_in,
                              void* d_out, int out_size, void* d_ws, size_t ws_size,
                              hipStream_t stream) {
    const float* x    = (const float*)d_in[0];
    float*       out  = (float*)d_out;
    float*       davg = (float*)d_ws;   // 256 floats of scratch

    davg_kernel <<<NB * ND,   1024, 0, stream>>>(x, davg);
    scale_kernel<<<N_CHUNKS,   256, 0, stream>>>(x, davg, out);
}